// DCT2D_Layer_67448166416975
// MI455X (gfx1250) — compile-verified
//
#include <hip/hip_runtime.h>

// ---------------------------------------------------------------------------
// DCT-8x8 + zigzag feature map for MI455X (gfx1250, wave32, WMMA).
//
// img: [16, 3, 512, 512] f32   ->   out: [16, 3*64, 64, 64] f32
//
// Strategy:
//  * 16x16 pixel tile = 2x2 grid of 8x8 DCT blocks.
//  * A = diag(C8, C8) block-diagonal DCT-II basis (16x16).
//    Z = A * X * A^T  computes all four 8x8 DCTs of the tile at once.
//  * Each 16x16x16 product = 4 chained V_WMMA_F32_16X16X4_F32 (f32 in/out,
//    K=4), accumulator carried in v8f across the chain.
//  * Layout symmetry: the per-lane register image of A as SRC0 (A-matrix,
//    16x4 chunks: M=lane%16, K=vgpr+2*(lane>>4)) equals the image of A^T as
//    SRC1 (B-matrix, 4x16 chunks: N=lane%16, K=vgpr+2*(lane>>4)), so one
//    constant operand set feeds both passes.
//  * Pass-1 result (C/D layout: M=vgpr+8*(lane>>4), N=lane%16) is bounced
//    through wave-private padded LDS to re-emerge in A-operand layout.
//    DS ops are in-order within a wave -> no barrier needed for this bounce.
//  * Zigzag is applied on the SCATTER side: coefficients land in the staging
//    buffer at their inverse-zigzag slot, so the coalesced writer loop is a
//    pure linear ds_load + global_store with no table lookups on the critical
//    path. The 8 IZZ values each lane needs are tile-invariant and hoisted.
// ---------------------------------------------------------------------------

typedef float v2f __attribute__((ext_vector_type(2)));
typedef float v8f __attribute__((ext_vector_type(8)));

#define IMG_W 512
#define IMG_H 512

// Inverse zigzag: IZZ[u*8+v] = zigzag slot of raster coefficient (u,v).
__device__ __constant__ int IZZ[64] = {
     0,  1,  5,  6, 14, 15, 27, 28,
     2,  4,  7, 13, 16, 26, 29, 42,
     3,  8, 12, 17, 25, 30, 41, 43,
     9, 11, 18, 24, 31, 40, 44, 53,
    10, 19, 23, 32, 39, 45, 52, 54,
    20, 22, 33, 38, 46, 51, 55, 60,
    21, 34, 37, 47, 50, 56, 59, 61,
    35, 36, 48, 49, 57, 58, 62, 63};

__global__ __launch_bounds__(256) void dct8x8_zigzag_kernel(
    const float* __restrict__ img, float* __restrict__ out)
{
    // Wave-private bounce buffer for pass1->pass2 relayout (16x16, pad to 17).
    __shared__ float ysc[8][16][17];          //  8.5 KB
    // Tile-row staging: [bi(2)][mj(64)][zigzag slot(64) padded to 65].
    __shared__ float stage[2 * 64 * 65];      // 32.5 KB

    const int lane = threadIdx.x & 31;
    const int wave = threadIdx.x >> 5;
    const int ti   = blockIdx.x;              // tile row   0..31
    const int bc   = blockIdx.y;              // b*3 + c    0..47

    const int lrow = lane & 15;               // M (pass A/B: row / col index)
    const int lhi  = lane >> 4;               // upper lane half

    // Constant operand: block-diagonal DCT basis, chunked K=4.
    // aop[q][j] = A[lrow][4q + j + 2*lhi],
    //   A[u][r] = (u/8==r/8) ? 2*cos(pi*(2*(r%8)+1)*(u%8)/16) : 0
    v2f aop[4];
#pragma unroll
    for (int q = 0; q < 4; ++q) {
#pragma unroll
        for (int j = 0; j < 2; ++j) {
            const int u = lrow;
            const int r = 4 * q + j + 2 * lhi;
            float val = 0.0f;
            if ((u >> 3) == (r >> 3)) {
                val = 2.0f * __cosf(3.14159265358979323846f *
                                    (2.0f * (float)(r & 7) + 1.0f) *
                                    (float)(u & 7) * (1.0f / 16.0f));
            }
            aop[q][j] = val;
        }
    }

    // Hoisted inverse-zigzag column for this lane's spatial column vv=lrow&7:
    // scatter slot for coefficient row u (== Z accumulator vgpr index).
    int izzc[8];
#pragma unroll
    for (int u = 0; u < 8; ++u) {
        izzc[u] = IZZ[u * 8 + (lrow & 7)];
    }

    const float* src = img + (size_t)bc * (IMG_H * IMG_W);

    // Each wave handles 4 of the 32 tiles in this tile row.
#pragma unroll 1
    for (int it = 0; it < 4; ++it) {
        const int tj = wave * 4 + it;

        // Load X tile directly in B-operand layout (K chunks of 4 rows).
        v2f bx[4];
#pragma unroll
        for (int q = 0; q < 4; ++q) {
#pragma unroll
            for (int j = 0; j < 2; ++j) {
                const int row = ti * 16 + 4 * q + j + 2 * lhi;
                const int col = tj * 16 + lrow;
                bx[q][j] = src[row * IMG_W + col];
            }
        }

        // Pass 1: Y = A * X   (4 chained f32 WMMAs, K=4 each)
        v8f acc = {};
#pragma unroll
        for (int q = 0; q < 4; ++q) {
            acc = __builtin_amdgcn_wmma_f32_16x16x4_f32(
                false, aop[q], false, bx[q], (short)0, acc, false, false);
        }

        // C/D-layout -> LDS (wave-private; DS in-order per wave).
#pragma unroll
        for (int v = 0; v < 8; ++v) {
            ysc[wave][v + 8 * lhi][lrow] = acc[v];
        }

        // Pass 2: Z = Y * A^T. Reload Y in A-operand layout; aop doubles as
        // the A^T B-operand (identical per-lane register image).
        v8f acc2 = {};
#pragma unroll
        for (int q = 0; q < 4; ++q) {
            v2f yop;
#pragma unroll
            for (int j = 0; j < 2; ++j) {
                yop[j] = ysc[wave][lrow][4 * q + j + 2 * lhi];
            }
            acc2 = __builtin_amdgcn_wmma_f32_16x16x4_f32(
                false, yop, false, aop[q], (short)0, acc2, false, false);
        }

        // Scatter Z into the tile-row staging buffer at its zigzag slot:
        // stage[bi][mj][IZZ[u*8+vv]]  (slot table hoisted to izzc[]).
#pragma unroll
        for (int v = 0; v < 8; ++v) {
            const int m  = v + 8 * lhi;       // tile row of coeff (u = v)
            const int n  = lrow;              // tile col of coeff
            const int bi = m >> 3;
            const int bj = n >> 3;
            const int mj = tj * 2 + bj;
            stage[(bi * 64 + mj) * 65 + izzc[v]] = acc2[v];
        }
    }

    __syncthreads();

    // Coalesced writer: 128 output rows (bi in {0,1}, k in 0..63), each a
    // contiguous 64-float (256 B) segment of out[(bc*64+k)][mi][*].
    // Pure linear ds_load + global_store; stride-65 reads are conflict-free.
    float* dst = out + (size_t)bc * 64 * 64 * 64;
    const int mj    = threadIdx.x & 63;
    const int rbase = threadIdx.x >> 6;       // 0..3
#pragma unroll 4
    for (int itr = 0; itr < 32; ++itr) {
        const int row = itr * 4 + rbase;      // 0..127
        const int k   = row & 63;
        const int bi  = row >> 6;
        const float v = stage[(bi * 64 + mj) * 65 + k];
        dst[(size_t)k * 4096 + (ti * 2 + bi) * 64 + mj] = v;
    }
}

extern "C" void kernel_launch(void* const* d_in, const int* in_sizes, int n_in,
                              void* d_out, int out_size, void* d_ws, size_t ws_size,
                              hipStream_t stream) {
    const float* img = (const float*)d_in[0];
    float* out = (float*)d_out;

    const int planes = in_sizes[0] / (IMG_H * IMG_W);   // bs*c = 48
    dim3 grid(IMG_H / 16, planes, 1);                   // (32, 48)
    dct8x8_zigzag_kernel<<<grid, 256, 0, stream>>>(img, out);
}